// SoftDiscretizationEncoder_27298812133418
// MI455X (gfx1250) — compile-verified
//
#include <hip/hip_runtime.h>

// CDNA5 / gfx1250, wave32.
// Out[B,768] = W[B,20] x E[20,768]; W sparse (2 nnz/row) from searchsorted +
// interpolation alpha. Store-bandwidth bound (~403 MB out @ 23.3 TB/s => ~17us
// floor). Embed table (61 KB) staged in LDS (async-to-LDS when available),
// math via V_WMMA_F32_16X16X4_F32, output via non-temporal stores.

typedef __attribute__((ext_vector_type(2))) float v2f;
typedef __attribute__((ext_vector_type(8))) float v8f;
typedef int v4i __attribute__((vector_size(16)));   // builtin's pointee type

#define NUM_BINS 20
#define NB1 19                   // boundaries per code row
#define EMBED 768
#define COL_TILES (EMBED / 16)   // 48
#define EPS_F 1e-8f

template<bool FULL>
__global__ __launch_bounds__(256) void soft_disc_wmma_kernel(
    const float* __restrict__ values,
    const int*   __restrict__ code_ids,
    const float* __restrict__ boundaries,
    const float* __restrict__ embed,     // [20, 768] row-major
    float*       __restrict__ out,       // [batch, 768]
    int batch, int row_base)
{
    // ---- Stage embed table into LDS (20*768*4 = 61440 B) ----
    __shared__ float ldsE[NUM_BINS * EMBED];

#if __has_builtin(__builtin_amdgcn_global_load_async_to_lds_b128)
    for (int i = threadIdx.x; i < (NUM_BINS * EMBED) / 4; i += 256) {
        __builtin_amdgcn_global_load_async_to_lds_b128(
            (v4i*)(embed + i * 4),   // global source
            (v4i*)(ldsE + i * 4),    // LDS destination
            /*imm offset=*/0, /*cpol=*/0);
    }
  #if __has_builtin(__builtin_amdgcn_s_wait_asynccnt)
    __builtin_amdgcn_s_wait_asynccnt(0);
  #else
    asm volatile("s_wait_asynccnt 0" ::: "memory");
  #endif
#else
    {
        const float4* src = (const float4*)embed;
        float4*       dst = (float4*)ldsE;
        for (int i = threadIdx.x; i < (NUM_BINS * EMBED) / 4; i += 256)
            dst[i] = src[i];
    }
#endif
    __syncthreads();

    const int lane = threadIdx.x & 31;
    const int wave = threadIdx.x >> 5;
    const int row0 = row_base + (blockIdx.x * 8 + wave) * 16;  // 16 rows/tile
    if (!FULL && row0 >= batch) return;

    const int m    = lane & 15;        // matrix row this lane computes
    const int koff = (lane >> 4) * 2;  // 0 for lanes 0-15, 2 for lanes 16-31

    // ---- Per-row scalar math (duplicated across both half-waves) ----
    int row = row0 + m;
    if (!FULL && row > batch - 1) row = batch - 1;   // clamp for tail tile

    const float v    = values[row];
    const int   code = code_ids[row];
    const float* brow = boundaries + (size_t)code * NB1;

    // searchsorted(side='left'): idx = #(b[j] < v); capture lo_b = b[idx-1],
    // hi_b = b[idx] on the fly (boundaries sorted) -> no dynamic indexing.
    int   idx  = 0;
    float lo_b = 0.0f, hi_b = 0.0f;
    bool  hset = false;
    #pragma unroll
    for (int j = 0; j < NB1; ++j) {
        float bj = brow[j];
        if (bj < v) { idx++; lo_b = bj; }
        else if (!hset) { hi_b = bj; hset = true; }
    }
    if (idx == 0)   lo_b = hi_b;   // b[clip(idx-1,0,18)] == b[0]
    if (idx == NB1) hi_b = lo_b;   // b[clip(idx,0,18)]   == b[18]

    const bool interior = (idx > 0) && (idx < NB1);
    const float denom = hi_b - lo_b;
    const bool  tiny  = fabsf(denom) < EPS_F;
    float acl   = fminf(fmaxf((v - lo_b) / denom, 0.0f), 1.0f);
    float alpha = tiny ? 0.5f : acl;
    alpha       = interior ? alpha : 0.0f;

    const int lower = (idx == 0) ? 0 : ((idx >= NB1) ? NB1 : idx - 1);
    const int upper = interior ? idx : lower;
    const float wl = 1.0f - alpha;
    const float wu = alpha;

    // ---- A fragments: W tile 16x20 as five 16x4 f32 WMMA operands.
    // f32 A 16x4 layout: lanes 0-15 hold K = {4kc, 4kc+1} for M = lane;
    // lanes 16-31 hold K = {4kc+2, 4kc+3} for M = lane-16.
    v2f afrag[5];
    #pragma unroll
    for (int kc = 0; kc < 5; ++kc) {
        int k0 = kc * 4 + koff;
        afrag[kc].x = ((k0     == lower) ? wl : 0.0f) + ((k0     == upper) ? wu : 0.0f);
        afrag[kc].y = ((k0 + 1 == lower) ? wl : 0.0f) + ((k0 + 1 == upper) ? wu : 0.0f);
    }

    // ---- Sweep 48 column tiles: D(16x16) = sum_kc A[kc](16x4) * B[kc](4x16)
    const int colL = lane & 15;
    const int mhi  = (lane >> 4) * 8;   // D rows: r (lanes 0-15) / r+8 (16-31)

    for (int nt = 0; nt < COL_TILES; ++nt) {
        const int col = nt * 16 + colL;

        v8f c = {};
        #pragma unroll
        for (int kc = 0; kc < 5; ++kc) {
            int k0 = kc * 4 + koff;
            // f32 B 4x16 layout mirrors A: lanes 0-15 -> K = {4kc, 4kc+1},
            // lanes 16-31 -> K = {4kc+2, 4kc+3}, N = lane & 15.
            v2f bfrag;
            bfrag.x = ldsE[(k0    ) * EMBED + col];
            bfrag.y = ldsE[(k0 + 1) * EMBED + col];
            c = __builtin_amdgcn_wmma_f32_16x16x4_f32(
                    /*neg_a=*/false, afrag[kc],
                    /*neg_b=*/false, bfrag,
                    /*c_mod=*/(short)0, c,
                    /*reuse_a=*/false, /*reuse_b=*/false);
        }

        // D layout: VGPR r holds M = r (lanes 0-15) / M = r+8 (lanes 16-31),
        // N = lane & 15.
        if (FULL) {
            // Branch-free streaming stores: base + immediate row offsets.
            float* outp = out + (size_t)(row0 + mhi) * EMBED + col;
            #pragma unroll
            for (int r = 0; r < 8; ++r)
                __builtin_nontemporal_store(c[r], outp + (size_t)r * EMBED);
        } else {
            #pragma unroll
            for (int r = 0; r < 8; ++r) {
                int rr = r + mhi;
                if (row0 + rr < batch)
                    out[(size_t)(row0 + rr) * EMBED + col] = c[r];
            }
        }
    }
}

extern "C" void kernel_launch(void* const* d_in, const int* in_sizes, int n_in,
                              void* d_out, int out_size, void* d_ws, size_t ws_size,
                              hipStream_t stream) {
    const float* values     = (const float*)d_in[0];
    const int*   code_ids   = (const int*)  d_in[1];
    const float* boundaries = (const float*)d_in[2];
    const float* embed      = (const float*)d_in[3];
    float*       out        = (float*)d_out;

    const int batch      = in_sizes[0];
    const int fullBlocks = batch / 128;          // 8 waves x 16 rows per block
    const int rem        = batch - fullBlocks * 128;

    if (fullBlocks > 0)
        soft_disc_wmma_kernel<true><<<fullBlocks, 256, 0, stream>>>(
            values, code_ids, boundaries, embed, out, batch, 0);
    if (rem > 0)
        soft_disc_wmma_kernel<false><<<1, 256, 0, stream>>>(
            values, code_ids, boundaries, embed, out, batch, fullBlocks * 128);
}